// CRF_84902913507725
// MI455X (gfx1250) — compile-verified
//
#include <hip/hip_runtime.h>
#include <stdint.h>

// Problem shape (fixed by reference)
#define Bn 64
#define Tn 2048
#define Sn 128
#define SUBS 4                      // j-chunks per state, adjacent lanes in a wave
#define JPC (Sn / SUBS)             // 32 j's per lane
#define NTHREADS (Sn * SUBS)        // 512 threads = 16 wave32
#define EMSLOTS 8                   // async emission pipeline depth (covers HBM latency)

// Dynamic LDS layout (bytes):
//   t1buf  : 2*128 f32        (double-buffered T1)
//   embuf  : 8*128 f32        (8-deep rotating emission buffer, async-to-LDS)
//   zbuf   : 2048 u8          (decoded path)
//   bp     : 2047*128 u8      (ALL backpointers — fits in CDNA5 320KB WGP LDS)
#define SMEM_BYTES ((2*Sn + EMSLOTS*Sn)*4 + Tn + (Tn-1)*Sn)

extern "C" __global__ __launch_bounds__(NTHREADS)
void viterbi_crf_gfx1250_v2(const float* __restrict__ trans,
                            const float* __restrict__ emis,
                            const int* __restrict__ lens,
                            float* __restrict__ out)
{
  extern __shared__ unsigned char smem[];
  float* t1buf = (float*)smem;                               // [2][128]
  float* embuf = t1buf + 2 * Sn;                             // [EMSLOTS][128]
  unsigned char* zbuf = (unsigned char*)(embuf + EMSLOTS*Sn);// [2048]
  unsigned char* bp   = zbuf + Tn;                           // [2047][128]

  const int n   = blockIdx.x;
  const int tid = threadIdx.x;
  const int sub = tid & (SUBS - 1);   // j-chunk (lanes 4k..4k+3 share state)
  const int i   = tid >> 2;           // owned state 0..127
  const int j0  = sub * JPC;
  const int len = lens[n];

  const float* trN = trans + (size_t)n * (Sn + 1) * Sn;
  const float* emN = emis  + (size_t)n * Tn * Sn;

  // This lane's transition slice in 32 VGPRs: tr[jj] = trans[n][j0+jj][i]
  float tr[JPC];
  #pragma unroll
  for (int jj = 0; jj < JPC; ++jj)
    tr[jj] = trN[(size_t)(j0 + jj) * Sn + i];

  // t = 0 init: T1_0[i] = trans[n][S][i] + em[n][0][i]
  if (sub == 0)
    t1buf[i] = trN[(size_t)Sn * Sn + i] + emN[i];

  // Prime the async pipeline: rows 1..7 into slots 1..7 (7 outstanding).
  // Note: each wave loads exactly the em values its own lanes consume,
  // so its own s_wait_asynccnt fully covers its reads.
  #pragma unroll
  for (int r = 1; r < EMSLOTS; ++r) {
    uint32_t ldsa = (uint32_t)(uintptr_t)&embuf[r * Sn + i];
    const float* g = emN + (size_t)r * Sn + i;
    asm volatile("global_load_async_to_lds_b32 %0, %1, off"
                 :: "v"(ldsa), "v"(g) : "memory");
  }
  __syncthreads();

  int p = 0;
  for (int t = 1; t < Tn; ++t) {
    // Issue row t+7 into slot (t+7)&7 (row clamped; slot rotation keeps the
    // wait-count invariant and never overwrites an unconsumed slot).
    {
      int rnext = t + EMSLOTS - 1; if (rnext > Tn - 1) rnext = Tn - 1;
      uint32_t ldsa =
        (uint32_t)(uintptr_t)&embuf[((t + EMSLOTS - 1) & (EMSLOTS - 1)) * Sn + i];
      const float* g = emN + (size_t)rnext * Sn + i;
      asm volatile("global_load_async_to_lds_b32 %0, %1, off"
                   :: "v"(ldsa), "v"(g) : "memory");
    }

    // Max-plus partial over this lane's 32-wide j-chunk (8x ds_load_b128).
    const float4* pv = (const float4*)&t1buf[p * Sn + j0];
    float best = -3.402823466e38f;
    int bj = j0;
    #pragma unroll
    for (int q = 0; q < JPC / 4; ++q) {
      float4 v = pv[q];
      float s0 = v.x + tr[4*q + 0];
      float s1 = v.y + tr[4*q + 1];
      float s2 = v.z + tr[4*q + 2];
      float s3 = v.w + tr[4*q + 3];
      if (s0 > best) { best = s0; bj = j0 + 4*q + 0; }  // strict > == first-max
      if (s1 > best) { best = s1; bj = j0 + 4*q + 1; }
      if (s2 > best) { best = s2; bj = j0 + 4*q + 2; }
      if (s3 > best) { best = s3; bj = j0 + 4*q + 3; }
    }

    // Intra-wave butterfly over the 4 sub-lanes of each state (xor 1, then 2).
    // Tie-break on smaller j to preserve jnp.argmax first-occurrence semantics.
    #pragma unroll
    for (int m = 1; m <= 2; m <<= 1) {
      float ov = __shfl_xor(best, m, 32);
      int   oj = __shfl_xor(bj,   m, 32);
      bool take = (ov > best) || ((ov == best) && (oj < bj));
      best = take ? ov : best;
      bj   = take ? oj : bj;
    }

    // Retire the transfer for row t (issued 7 iterations ago).
    asm volatile("s_wait_asynccnt 0x7" ::: "memory");

    if (sub == 0) {
      float t1n  = best + embuf[(t & (EMSLOTS - 1)) * Sn + i];
      float prev = t1buf[p * Sn + i];
      t1buf[(p ^ 1) * Sn + i] = (t < len) ? t1n : prev;   // freeze past length
      bp[(size_t)(t - 1) * Sn + i] = (unsigned char)bj;   // bp recorded always
    }
    __syncthreads();   // single barrier per step: orders the t1 buffer flip
    p ^= 1;
  }

  // Backtrack: serial, but every dependent read hits LDS (not HBM).
  if (tid == 0) {
    const float* t1f = &t1buf[p * Sn];
    int zlast = 0; float bv = t1f[0];
    for (int s = 1; s < Sn; ++s) {
      float v = t1f[s];
      if (v > bv) { bv = v; zlast = s; }   // first-max
    }
    int ptr = 0;
    for (int t = Tn - 1; t >= 0; --t) {
      if (t == len - 1) ptr = zlast;
      zbuf[t] = (t <= len - 1) ? (unsigned char)ptr : (unsigned char)0;
      if (t >= 1) ptr = bp[(size_t)(t - 1) * Sn + ptr];
    }
  }
  __syncthreads();

  // Coalesced write of the decoded path (int tags as exact floats).
  float* outN = out + (size_t)n * Tn;
  for (int t = tid; t < Tn; t += NTHREADS)
    outN[t] = (float)zbuf[t];
}

extern "C" void kernel_launch(void* const* d_in, const int* in_sizes, int n_in,
                              void* d_out, int out_size, void* d_ws, size_t ws_size,
                              hipStream_t stream) {
  (void)in_sizes; (void)n_in; (void)d_ws; (void)ws_size; (void)out_size;
  const float* trans = (const float*)d_in[0];   // [64][129][128] f32
  const float* emis  = (const float*)d_in[1];   // [64][2048][128] f32
  const int*   lens  = (const int*)d_in[2];     // [64] i32
  float* out = (float*)d_out;                   // [64][2048]
  viterbi_crf_gfx1250_v2<<<Bn, NTHREADS, SMEM_BYTES, stream>>>(trans, emis, lens, out);
}